// CtcLayer_55087250538885
// MI455X (gfx1250) — compile-verified
//
#include <hip/hip_runtime.h>
#include <hip/hip_bf16.h>

// Problem constants (match reference).
#define BB   512
#define TT   512
#define CC   128
#define LL   64
#define SS   (2 * LL + 1)   // 129 extended states
#define NEGF (-1e30f)
#define EPSF (1e-7f)

#define PD    4             // async prefetch distance (rows in flight)
#define NBUF  6             // rotating row buffers; NBUF >= PD + 2 for safety
#define ROWB  (CC * 4)      // 512 bytes per probability row

#define BLOCK_THREADS 160   // 5 wave32s; threads 0..128 own DP states

__global__ __launch_bounds__(BLOCK_THREADS)
void ctc_fwd_kernel(const int* __restrict__ y_true,   // [B, L] int32
                    const float* __restrict__ y_pred, // [B, T, C] float32
                    float* __restrict__ out) {        // [B, 1] float32
    // LDS: NBUF rotating probability rows, ping-pong alpha (+2 NEG pad), labels.
    __shared__ float rowbuf[NBUF][CC];     // 3072 B
    __shared__ float alpha[2][SS + 2];     // 1048 B
    __shared__ int   labs[LL];             // 256 B

    const int tid = threadIdx.x;
    const int b   = blockIdx.x;
    const float* rowbase = y_pred + (size_t)b * TT * CC;

    // Stage labels and NEG pads.
    if (tid < LL) labs[tid] = y_true[b * LL + tid];
    if (tid < 2) { alpha[0][tid] = NEGF; alpha[1][tid] = NEGF; }

    // 32-bit LDS byte address of the row buffers (wave-relative LDS offset).
    const unsigned ldsRow = (unsigned)(uintptr_t)(&rowbuf[0][0]);

    // Wave 0: prime the async pipeline with rows 0..PD-1 -> bufs 0..PD-1.
    // 32 lanes x 16B (B128) = 512B = one full C=128 row per issue.
    if (tid < 32) {
        const unsigned laneOff = (unsigned)tid * 16u;
        #pragma unroll
        for (int r = 0; r < PD; ++r) {
            unsigned lds = ldsRow + (unsigned)(r * ROWB) + laneOff;
            unsigned off = (unsigned)(r * ROWB) + laneOff;
            asm volatile("global_load_async_to_lds_b128 %0, %1, %2"
                         :: "v"(lds), "v"(off), "s"(rowbase) : "memory");
        }
    }

    __syncthreads();   // labels visible to all state threads

    // Per-thread DP-state metadata in registers.
    const int s = tid;
    int  extc = CC - 1;     // blank = C-1
    bool skip = false;
    if (s < SS && (s & 1)) {
        extc = labs[(s - 1) >> 1];
        if (s >= 3) skip = (labs[(s - 1) >> 1] != labs[(s - 3) >> 1]);
    }

    int rdBuf = 0;          // buffer holding row t
    int wrBuf = PD % NBUF;  // buffer to receive row t+PD

    for (int t = 0; t < TT; ++t) {
        // Wave 0: issue row min(t+PD, T-1) into wrBuf, then guarantee row t
        // has landed (async loads complete in order; PD+1 in flight max).
        if (tid < 32) {
            int r = t + PD;
            if (r > TT - 1) r = TT - 1;   // tail: redundant reload of last row
            unsigned lds = ldsRow + (unsigned)(wrBuf * ROWB) + (unsigned)tid * 16u;
            unsigned off = (unsigned)(r * ROWB) + (unsigned)tid * 16u;
            asm volatile("global_load_async_to_lds_b128 %0, %1, %2"
                         :: "v"(lds), "v"(off), "s"(rowbase) : "memory");
            asm volatile("s_wait_asynccnt 0x4" ::: "memory");   // row t complete
        }
        __syncthreads();   // row t visible; prev-iter alpha writes visible

        if (s < SS) {
            float lp = __logf(rowbuf[rdBuf][extc] + EPSF);
            float na;
            if (t == 0) {
                na = (s < 2) ? lp : NEGF;
            } else {
                const float* ap = alpha[(t - 1) & 1];
                float a0 = ap[s + 2];              // alpha[s]
                float a1 = ap[s + 1];              // alpha[s-1]
                float a2 = skip ? ap[s] : NEGF;    // alpha[s-2] if skip allowed
                // stable logaddexp3: m + log(sum exp(ai - m))
                float m  = fmaxf(fmaxf(a0, a1), a2);
                float sm = __expf(a0 - m) + __expf(a1 - m) + __expf(a2 - m);
                na = m + __logf(sm) + lp;
            }
            alpha[t & 1][s + 2] = na;
        }

        // rotate buffer indices (cheap conditional wrap, no integer division)
        rdBuf = (rdBuf == NBUF - 1) ? 0 : rdBuf + 1;
        wrBuf = (wrBuf == NBUF - 1) ? 0 : wrBuf + 1;
    }

    __syncthreads();
    if (tid == 0) {
        const float* ap = alpha[(TT - 1) & 1];
        float aN  = ap[SS + 1];   // state S-1 (last blank)
        float aN1 = ap[SS];       // state S-2 (last label)
        float m   = fmaxf(aN, aN1);
        float ll  = m + __logf(__expf(aN - m) + __expf(aN1 - m));
        out[b] = -ll;
    }
}

extern "C" void kernel_launch(void* const* d_in, const int* in_sizes, int n_in,
                              void* d_out, int out_size, void* d_ws, size_t ws_size,
                              hipStream_t stream) {
    (void)n_in; (void)d_ws; (void)ws_size; (void)out_size; (void)in_sizes;
    const int*   y_true = (const int*)d_in[0];    // [B, L] int32
    const float* y_pred = (const float*)d_in[1];  // [B, T, C] float32
    float*       out    = (float*)d_out;          // [B, 1] float32
    ctc_fwd_kernel<<<dim3(BB), dim3(BLOCK_THREADS), 0, stream>>>(y_true, y_pred, out);
}